// SpatialFeaturePreProcessingLayer_19945828123241
// MI455X (gfx1250) — compile-verified
//
#include <hip/hip_runtime.h>
#include <math.h>

// Problem geometry (fixed by the reference)
#define BB 64
#define FF 17
#define HH 128
#define WW 128
#define HWSZ (HH * WW)            // 16384
#define NPIX (BB * HWSZ)          // 1,048,576
#define TOTAL_SCALE 1914
#define N_CAT 9
#define BLOCK 128                 // 4 waves
#define VEC 4                     // pixels per thread
#define PIX_PER_BLOCK (BLOCK * VEC)   // 512
#define PIX_PER_WAVE  (32 * VEC)      // 128
#define WAVE_DWORDS   (PIX_PER_WAVE * FF)  // 2176 = 17 * 128 exactly

// Per-feature: -1 => scalar (log(t+1)), else categorical index
__device__ constexpr int kCatIdx[FF] = {
    -1,  // height_map
     0,  // visibility_map
     1,  // creep
     2,  // power
     3,  // player_id
     4,  // player_relative
     5,  // unit_type
     6,  // selected
    -1, -1, -1, -1, -1, -1,  // hit_points..shields_ratio (scalar)
     7,  // unit_density
    -1,  // unit_density_aa
     8   // effects
};
__device__ constexpr int kCatOff[N_CAT] = {0, 4, 6, 8, 25, 30, 1880, 1882, 1898};

__global__ __launch_bounds__(BLOCK)
void spp_kernel(const int* __restrict__ feat,
                const float* __restrict__ weights,
                const float* __restrict__ biases,
                float* __restrict__ out)
{
    __shared__ float s_w[TOTAL_SCALE];              // embedding table, 7656 B
    __shared__ float s_t[PIX_PER_BLOCK * FF];       // transpose buffer, 34816 B

    const int tid  = threadIdx.x;
    const int lane = tid & 31;
    const int wave = tid >> 5;

    // ---- Stage weight table into LDS via gfx1250 async global->LDS DMA ----
    // 478 x b128 chunks (dwords 0..1911), then a 2-dword b32 tail.
    for (int i = tid; i < TOTAL_SCALE / 4; i += BLOCK) {   // 478 chunks
        unsigned int ldsOff = (unsigned int)(unsigned long long)(&s_w[i * 4]);
        unsigned int gOff   = (unsigned int)(i * 16);
        asm volatile("global_load_async_to_lds_b128 %0, %1, %2"
                     :: "v"(ldsOff), "v"(gOff), "s"(weights)
                     : "memory");
    }
    if (tid < (TOTAL_SCALE & 3)) {                         // dwords 1912, 1913
        const int i = (TOTAL_SCALE & ~3) + tid;
        unsigned int ldsOff = (unsigned int)(unsigned long long)(&s_w[i]);
        unsigned int gOff   = (unsigned int)(i * 4);
        asm volatile("global_load_async_to_lds_b32 %0, %1, %2"
                     :: "v"(ldsOff), "v"(gOff), "s"(weights)
                     : "memory");
    }
    asm volatile("s_wait_asynccnt 0" ::: "memory");
    __syncthreads();

    // ---- 4 consecutive pixels per thread ----
    const int p0  = blockIdx.x * PIX_PER_BLOCK + tid * VEC;  // first pixel
    const int b   = p0 >> 14;                                // p0 / 16384
    const int hw0 = p0 & (HWSZ - 1);
    const long long inBase = (long long)b * FF * HWSZ + hw0;

    float* tbuf = &s_t[wave * WAVE_DWORDS];
    const int pw = lane * VEC;                               // pixel-in-wave of j==0

#pragma unroll
    for (int f = 0; f < FF; ++f) {
        // one b128 load covers this channel for all 4 pixels (16B aligned, coalesced)
        const int4 tv = *(const int4*)(feat + inBase + (long long)f * HWSZ);
        const int tj[VEC] = {tv.x, tv.y, tv.z, tv.w};
        const int c = kCatIdx[f];                            // compile-time after unroll
#pragma unroll
        for (int j = 0; j < VEC; ++j) {
            float v;
            if (c < 0) {
                v = __logf((float)tj[j] + 1.0f);
            } else {
                v = s_w[kCatOff[c] + tj[j]] + biases[c];     // LDS gather + uniform bias
            }
            tbuf[(pw + j) * FF + f] = v;                     // scatter, stride 17
        }
    }
    __syncthreads();

    // ---- Coalesced drain: 17 rounds of 128 dwords per wave, b128 each ----
    float* outBase = out + (long long)(blockIdx.x * PIX_PER_BLOCK + wave * PIX_PER_WAVE) * FF;
#pragma unroll
    for (int r = 0; r < FF; ++r) {
        const int d = r * 128 + lane * 4;                    // 16B aligned
        const float4 v = *(const float4*)&tbuf[d];
        *(float4*)&outBase[d] = v;                           // 512 B coalesced store
    }
}

extern "C" void kernel_launch(void* const* d_in, const int* in_sizes, int n_in,
                              void* d_out, int out_size, void* d_ws, size_t ws_size,
                              hipStream_t stream)
{
    const int*   feat    = (const int*)d_in[0];
    const float* weights = (const float*)d_in[1];
    const float* biases  = (const float*)d_in[2];
    float*       out     = (float*)d_out;

    dim3 grid(NPIX / PIX_PER_BLOCK);   // 2048 blocks
    dim3 block(BLOCK);
    spp_kernel<<<grid, block, 0, stream>>>(feat, weights, biases, out);
}